// SparseGRN_47837345743025
// MI455X (gfx1250) — compile-verified
//
#include <hip/hip_runtime.h>

typedef __attribute__((ext_vector_type(2))) float v2f;
typedef __attribute__((ext_vector_type(4))) float v4f;
typedef __attribute__((ext_vector_type(8))) float v8f;

#define GRN_EPS 1e-6f

// x: (2,64,64,64,96) f32, gamma/beta: (1,96) f32, mask: (2,1,32,32,32) i32
// out[b,h,w,d,c] = (gamma[c]*x*Nx + beta[c]) * mask[b,0,h/2,w/2,d/2]
// Nx = Gx/(Gx+eps), Gx = sqrt(sum_c x^2)  (per voxel)
//
// One wave32 handles 16 consecutive voxels (16*96 floats = 6 KB).
// Lane L (<16): voxel L, channels 0..47 ; lane L+16: voxel L, channels 48..95.
// Per-lane partial sums of squares land in the 16x4 f32 A-matrix layout;
// one v_wmma_f32_16x16x4_f32 against an all-ones B reduces all 16 voxels.
__global__ __launch_bounds__(256) void sparse_grn_kernel(
    const float* __restrict__ x,
    const float* __restrict__ gamma,
    const float* __restrict__ beta,
    const int*   __restrict__ mask,
    float*       __restrict__ out)
{
    const int lane  = threadIdx.x & 31;
    const int wave  = threadIdx.x >> 5;
    const int group = blockIdx.x * 8 + wave;   // 16 voxels per group

    const int m16  = lane & 15;   // voxel within group
    const int half = lane >> 4;   // 0: ch 0..47, 1: ch 48..95

    // voxel linear index over (b,h,w,d), d fastest
    const int v = group * 16 + m16;
    const int d = v & 63;
    const int w = (v >> 6) & 63;
    const int h = (v >> 12) & 63;
    const int b = v >> 18;

    // mask: (B,1,32,32,32), nearest-neighbor 2x upsample
    const int midx = (((b * 32 + (h >> 1)) * 32 + (w >> 1)) * 32 + (d >> 1));
    const float mval = (mask[midx] != 0) ? 1.0f : 0.0f;

    const size_t base = (size_t)v * 96 + (size_t)half * 48;
    const v4f* __restrict__ xp = reinterpret_cast<const v4f*>(x + base);

    // Load 48 channels per lane; masked voxels skip the reads entirely
    // (EXEC-masked vmem => no HBM traffic for ~50% of voxels).
    v4f xv[12];
    if (mval != 0.0f) {
#pragma unroll
        for (int i = 0; i < 12; ++i)
            xv[i] = __builtin_nontemporal_load(xp + i);
    } else {
#pragma unroll
        for (int i = 0; i < 12; ++i)
            xv[i] = (v4f){0.0f, 0.0f, 0.0f, 0.0f};
    }

    // Two partial sums of squares per lane -> A[m][k] in the 16x4 f32 layout:
    //   lanes 0-15 : VGPR0=K0 (ch 0-23),  VGPR1=K1 (ch 24-47)
    //   lanes16-31 : VGPR0=K2 (ch 48-71), VGPR1=K3 (ch 72-95)
    float a0 = 0.0f, a1 = 0.0f;
#pragma unroll
    for (int i = 0; i < 6; ++i)
        a0 += xv[i][0] * xv[i][0] + xv[i][1] * xv[i][1] +
              xv[i][2] * xv[i][2] + xv[i][3] * xv[i][3];
#pragma unroll
    for (int i = 6; i < 12; ++i)
        a1 += xv[i][0] * xv[i][0] + xv[i][1] * xv[i][1] +
              xv[i][2] * xv[i][2] + xv[i][3] * xv[i][3];

    v2f A; A[0] = a0; A[1] = a1;
    v2f B; B[0] = 1.0f; B[1] = 1.0f;   // all-ones B => row sums of A
    v8f C = {};
    // D[r][n] = Gx^2 of voxel r (lanes 0-15: M=r, lanes 16-31: M=r+8)
    v8f D = __builtin_amdgcn_wmma_f32_16x16x4_f32(
        /*neg_a=*/false, A, /*neg_b=*/false, B,
        /*c_mod=*/(short)0, C, /*reuse_a=*/false, /*reuse_b=*/false);

    // Route voxel (lane&15)'s sum back to this lane:
    // wanted VGPR index = lane&7; own copy holds M=(lane&7)+8*bit4(lane),
    // we want M=(lane&7)+8*bit3(lane) -> take partner (xor 16) iff bit3^bit4.
    const int i3 = lane & 7;
    const float s0 = (i3 & 1) ? D[1] : D[0];
    const float s1 = (i3 & 1) ? D[3] : D[2];
    const float s2 = (i3 & 1) ? D[5] : D[4];
    const float s3 = (i3 & 1) ? D[7] : D[6];
    const float t0 = (i3 & 2) ? s1 : s0;
    const float t1 = (i3 & 2) ? s3 : s2;
    const float own  = (i3 & 4) ? t1 : t0;
    const float part = __shfl_xor(own, 16, 32);
    const float sumsq = (((lane >> 3) ^ (lane >> 4)) & 1) ? part : own;

    const float Gx = sqrtf(sumsq);
    const float scale = (Gx / (Gx + GRN_EPS)) * mval;   // Nx * m
    // (gamma*x*Nx + beta)*m == gamma*x*(Nx*m) + beta*m  for m in {0,1}

    v4f* __restrict__ op = reinterpret_cast<v4f*>(out + base);
    const v4f* __restrict__ gp = reinterpret_cast<const v4f*>(gamma + half * 48);
    const v4f* __restrict__ bp = reinterpret_cast<const v4f*>(beta  + half * 48);

#pragma unroll
    for (int i = 0; i < 12; ++i) {
        const v4f g  = gp[i];   // 768B of gamma/beta: permanently cache-hot
        const v4f bt = bp[i];
        v4f o;
        o[0] = fmaf(g[0] * xv[i][0], scale, bt[0] * mval);
        o[1] = fmaf(g[1] * xv[i][1], scale, bt[1] * mval);
        o[2] = fmaf(g[2] * xv[i][2], scale, bt[2] * mval);
        o[3] = fmaf(g[3] * xv[i][3], scale, bt[3] * mval);
        __builtin_nontemporal_store(o, op + i);
    }
}

extern "C" void kernel_launch(void* const* d_in, const int* in_sizes, int n_in,
                              void* d_out, int out_size, void* d_ws, size_t ws_size,
                              hipStream_t stream) {
    const float* x     = (const float*)d_in[0];
    const float* gamma = (const float*)d_in[1];
    const float* beta  = (const float*)d_in[2];
    const int*   mask  = (const int*)d_in[3];
    float* out = (float*)d_out;

    // B=2, H=W=D=64 -> 524288 voxels, 16 voxels/wave, 8 waves/block
    const int voxels = 2 * 64 * 64 * 64;
    const int groups = voxels / 16;      // 32768 waves
    const int blocks = groups / 8;       // 4096 blocks of 256 threads
    sparse_grn_kernel<<<blocks, 256, 0, stream>>>(x, gamma, beta, mask, out);
}